// AttentionPool_77738908057666
// MI455X (gfx1250) — compile-verified
//
#include <hip/hip_runtime.h>

// AttentionPool on MI455X (gfx1250, wave32).
// 16 sorted segments (graphs) per 256-thread block:
//   Phase 0: binary-search the 17 segment boundaries.
//   Phase 1: gate logits (nodes @ W + b), coalesced float4 loads + wave32
//            butterfly reduction; logits cached in LDS.
//   Phase 2: per-graph segment softmax in LDS; pack {weight, graph-id} float2;
//            pad 32 sentinel entries {0,-1} so the tail needs no bounds checks.
//   Phase 3: weighted pooling as fp32 WMMA: D(16 feat x 16 graphs) +=
//            A(16 feat x 4 nodes) * B(4 nodes x 16 graphs). Main loop handles
//            only full 4-node groups (zero predication); one tail group per
//            wave uses the LDS sentinels + N-clamped A rows. Exact fp32 via
//            V_WMMA_F32_16X16X4_F32; A scalars staged to encourage one big
//            global-load clause per group.
// Memory-bound: ~1 HBM pass over nodes (1.02 GB); second read hits L2.

typedef __attribute__((ext_vector_type(2))) float v2f;
typedef __attribute__((ext_vector_type(8))) float v8f;

#define GPB      16     // graphs per block
#define CAP      6144   // LDS-cached nodes per block (mean ~1953, ~95 sigma headroom)
#define NTHREADS 256
#define NWAVES   8

__device__ __forceinline__ int lower_bound_ll(const long long* g, int n, long long t) {
  int lo = 0, hi = n;
  while (lo < hi) { int mid = (lo + hi) >> 1; if (g[mid] < t) lo = mid + 1; else hi = mid; }
  return lo;
}

// dot(row[0:128], W) + bias, result broadcast to all 32 lanes of the wave
__device__ __forceinline__ float wave_dot128(const float* __restrict__ row,
                                             float4 wv, float bias, int lane) {
  const float4 x = *(const float4*)(row + 4 * lane);
  float p = x.x * wv.x + x.y * wv.y + x.z * wv.z + x.w * wv.w;
#pragma unroll
  for (int o = 16; o > 0; o >>= 1) p += __shfl_xor(p, o, 32);
  return p + bias;
}

__global__ __launch_bounds__(NTHREADS)
void attnpool_wmma_kernel(const float* __restrict__ nodes,
                          const long long* __restrict__ gidx,
                          const float* __restrict__ W,
                          const float* __restrict__ bptr,
                          float* __restrict__ out,
                          int N, int NG) {
  __shared__ float2 s_wg[CAP + 32];     // .x: logit->weight, .y: graph id (pad: {0,-1})
  __shared__ float  s_out[GPB * 128];   // pooled output accumulator
  __shared__ int    s_bounds[GPB + 1];
  __shared__ float  s_m[GPB], s_s[GPB];
  __shared__ float  s_red[NWAVES];

  const int tid   = threadIdx.x;
  const int lane  = tid & 31;
  const int wave  = tid >> 5;
  const int gbase = blockIdx.x * GPB;
  if (gbase >= NG) return;

  for (int i = tid; i < GPB * 128; i += NTHREADS) s_out[i] = 0.f;
  if (tid <= GPB)
    s_bounds[tid] = lower_bound_ll(gidx, N, (long long)(gbase + tid));
  __syncthreads();

  const int r0 = s_bounds[0], r1 = s_bounds[GPB];
  const int range = r1 - r0;
  const float4 wv  = ((const float4*)W)[lane];
  const float bias = bptr[0];

  if (range <= CAP) {
    // ---- Phase 1: logits into LDS (one node per wave per iteration) ----
    for (int j = r0 + wave; j < r1; j += NWAVES) {
      if (j + NWAVES < r1)
        __builtin_prefetch(nodes + (size_t)(j + NWAVES) * 128 + 4 * lane, 0, 3);
      const float l = wave_dot128(nodes + (size_t)j * 128, wv, bias, lane);
      if (lane == 0) s_wg[j - r0].x = l;
    }
    __syncthreads();

    // ---- Phase 2a: per-graph max and exp-sum (one wave per graph) ----
    for (int gg = wave; gg < GPB; gg += NWAVES) {
      const int lo = s_bounds[gg] - r0, hi = s_bounds[gg + 1] - r0;
      float m = -3.4e38f;
      for (int i = lo + lane; i < hi; i += 32) m = fmaxf(m, s_wg[i].x);
#pragma unroll
      for (int o = 16; o > 0; o >>= 1) m = fmaxf(m, __shfl_xor(m, o, 32));
      float s = 0.f;
      for (int i = lo + lane; i < hi; i += 32) s += __expf(s_wg[i].x - m);
#pragma unroll
      for (int o = 16; o > 0; o >>= 1) s += __shfl_xor(s, o, 32);
      if (lane == 0) { s_m[gg] = m; s_s[gg] = s; }
    }
    __syncthreads();

    // ---- Phase 2b: logits -> {normalized weight, graph id}; write sentinels
    for (int i = tid; i < range; i += NTHREADS) {
      const int g = (int)(gidx[r0 + i] - (long long)gbase);
      const float w = __expf(s_wg[i].x - s_m[g]) / (s_s[g] + 1e-8f);
      s_wg[i] = make_float2(w, (float)g);
    }
    if (tid < 32) s_wg[range + tid] = make_float2(0.f, -1.f);  // tail sentinels
    __syncthreads();

    // ---- Phase 3: WMMA pooling. A: 16 feat x 4 nodes, B: 4 nodes x 16 graphs
    const int half = lane >> 4;        // 0 -> K=0,1 ; 1 -> K=2,3
    const int lm   = lane & 15;        // A: feature row / B,D: graph column
    const float lmf = (float)lm;
    const int k0 = half * 2;
    v8f acc[8] = {};

    int j = r0 + wave * 4;
    // Main loop: full 4-node groups only -> zero bounds checks, no clamps.
    for (; j + 4 <= r1; j += NWAVES * 4) {
      const int i0 = j + k0 - r0;
      const float2 wg0 = s_wg[i0];
      const float2 wg1 = s_wg[i0 + 1];
      const float* rowp = nodes + (size_t)(j + k0) * 128 + lm;
      float ax[8], ay[8];
#pragma unroll
      for (int f = 0; f < 8; ++f) {
        ax[f] = rowp[f * 16];
        ay[f] = rowp[128 + f * 16];
      }
      v2f bb;
      bb.x = (wg0.y == lmf) ? wg0.x : 0.f;
      bb.y = (wg1.y == lmf) ? wg1.x : 0.f;
#pragma unroll
      for (int f = 0; f < 8; ++f) {
        v2f a; a.x = ax[f]; a.y = ay[f];
        acc[f] = __builtin_amdgcn_wmma_f32_16x16x4_f32(
            false, a, false, bb, (short)0, acc[f], false, false);
      }
    }
    // Tail: at most one partial group per wave. LDS sentinels make B loads
    // unconditional; A rows clamped to N-1 (annihilated by zero weights).
    if (j < r1) {
      const int j0 = j + k0, j1 = j + k0 + 1;
      const int ra0 = (j0 < N) ? j0 : (N - 1);
      const int ra1 = (j1 < N) ? j1 : (N - 1);
      const float2 wg0 = s_wg[j0 - r0];   // index <= range+2: sentinel region
      const float2 wg1 = s_wg[j1 - r0];
      const float* row0 = nodes + (size_t)ra0 * 128 + lm;
      const float* row1 = nodes + (size_t)ra1 * 128 + lm;
      float ax[8], ay[8];
#pragma unroll
      for (int f = 0; f < 8; ++f) {
        ax[f] = row0[f * 16];
        ay[f] = row1[f * 16];
      }
      v2f bb;
      bb.x = (wg0.y == lmf) ? wg0.x : 0.f;
      bb.y = (wg1.y == lmf) ? wg1.x : 0.f;
#pragma unroll
      for (int f = 0; f < 8; ++f) {
        v2f a; a.x = ax[f]; a.y = ay[f];
        acc[f] = __builtin_amdgcn_wmma_f32_16x16x4_f32(
            false, a, false, bb, (short)0, acc[f], false, false);
      }
    }

    // D[r][n]: lanes 0-15 -> M=r, lanes 16-31 -> M=r+8; column n = graph lm
#pragma unroll
    for (int f = 0; f < 8; ++f)
#pragma unroll
      for (int r = 0; r < 8; ++r)
        atomicAdd(&s_out[lm * 128 + f * 16 + half * 8 + r], acc[f][r]);
    __syncthreads();
  } else {
    // ---- Fallback (range > CAP, statistically unreachable): 3-pass recompute
    for (int g = 0; g < GPB; ++g) {
      const int lo = s_bounds[g], hi = s_bounds[g + 1];
      float m = -3.4e38f;
      for (int j = lo + wave; j < hi; j += NWAVES)
        m = fmaxf(m, wave_dot128(nodes + (size_t)j * 128, wv, bias, lane));
      if (lane == 0) s_red[wave] = m;
      __syncthreads();
      m = -3.4e38f;
#pragma unroll
      for (int k = 0; k < NWAVES; ++k) m = fmaxf(m, s_red[k]);
      __syncthreads();
      float s = 0.f;
      for (int j = lo + wave; j < hi; j += NWAVES)
        s += __expf(wave_dot128(nodes + (size_t)j * 128, wv, bias, lane) - m);
      if (lane == 0) s_red[wave] = s;
      __syncthreads();
      s = 0.f;
#pragma unroll
      for (int k = 0; k < NWAVES; ++k) s += s_red[k];
      __syncthreads();
      const float inv = 1.f / (s + 1e-8f);
      float4 a4 = {0.f, 0.f, 0.f, 0.f};
      for (int j = lo + wave; j < hi; j += NWAVES) {
        const float l = wave_dot128(nodes + (size_t)j * 128, wv, bias, lane);
        const float wgt = __expf(l - m) * inv;
        const float4 x = *(const float4*)(nodes + (size_t)j * 128 + 4 * lane);
        a4.x += wgt * x.x; a4.y += wgt * x.y; a4.z += wgt * x.z; a4.w += wgt * x.w;
      }
      atomicAdd(&s_out[g * 128 + 4 * lane + 0], a4.x);
      atomicAdd(&s_out[g * 128 + 4 * lane + 1], a4.y);
      atomicAdd(&s_out[g * 128 + 4 * lane + 2], a4.z);
      atomicAdd(&s_out[g * 128 + 4 * lane + 3], a4.w);
      __syncthreads();
    }
  }

  // ---- Write 16 graphs x 128 features, coalesced ----
  for (int i = tid; i < GPB * 128; i += NTHREADS)
    out[(size_t)gbase * 128 + i] = s_out[i];
}

extern "C" void kernel_launch(void* const* d_in, const int* in_sizes, int n_in,
                              void* d_out, int out_size, void* d_ws, size_t ws_size,
                              hipStream_t stream) {
  const float*     nodes = (const float*)d_in[0];
  const long long* gidx  = (const long long*)d_in[1];  // int64 per reference
  const float*     W     = (const float*)d_in[3];
  const float*     b     = (const float*)d_in[4];
  float*           out   = (float*)d_out;

  const int N  = in_sizes[0] / 128;
  const int NG = out_size / 128;
  const int nblocks = (NG + GPB - 1) / GPB;

  attnpool_wmma_kernel<<<dim3(nblocks), dim3(NTHREADS), 0, stream>>>(
      nodes, gidx, W, b, out, N, NG);
}